// dilated_iso_layer_53910429499428
// MI455X (gfx1250) — compile-verified
//
#include <hip/hip_runtime.h>

// ---------------------------------------------------------------------------
// Problem constants (from reference): B=2, C_IN=4, C_OUT=8, K=4, N=224, D=2
// new_k = 7, nH = 218, n_sub = 218*218 = 47524, n_flat = 2*47524 = 95048
// Feature vector per (sample, in-channel): 16 struct + 4 row + 4 col = 24
// Kernel matrix per in-channel: 24 x 192   (192 = C_OUT(8) * nPerms(24))
// ---------------------------------------------------------------------------
#define NH     218
#define NSUB   (NH * NH)          // 47524
#define NTOT   (2 * NSUB)         // 95048
#define NCOLS  192                // 8 outputs * 24 permutations
#define KF     24                 // feature length (GEMM K)
#define NKSTEP 6                  // 24 / 4
#define NTILE  12                 // 192 / 16
#define FDIM   128

typedef __attribute__((ext_vector_type(2))) float v2f;
typedef __attribute__((ext_vector_type(8))) float v8f;

// itertools.permutations(range(4)) in lexicographic order
__device__ __constant__ int PERM[24][4] = {
    {0,1,2,3},{0,1,3,2},{0,2,1,3},{0,2,3,1},{0,3,1,2},{0,3,2,1},
    {1,0,2,3},{1,0,3,2},{1,2,0,3},{1,2,3,0},{1,3,0,2},{1,3,2,0},
    {2,0,1,3},{2,0,3,1},{2,1,0,3},{2,1,3,0},{2,3,0,1},{2,3,1,0},
    {3,0,1,2},{3,0,2,1},{3,1,0,2},{3,1,2,0},{3,2,0,1},{3,2,1,0}};

// w[o,p,i,kk]: permuted kernel element for feature index kk (0..23)
__device__ __forceinline__ float wval(int o, int p, int i, int kk,
                                      const float* __restrict__ ks,
                                      const float* __restrict__ kr,
                                      const float* __restrict__ kc) {
    if (kk < 16) {
        int a = kk >> 2, b = kk & 3;
        // Kp[o,i,p,a,b] = k_struct[o,i,perm[p][a],perm[p][b]]
        return ks[((o * 4 + i) * 4 + PERM[p][a]) * 4 + PERM[p][b]];
    } else if (kk < 20) {
        return kr[(o * 4 + i) * 4 + PERM[p][kk - 16]];
    } else {
        return kc[(o * 4 + i) * 4 + PERM[p][kk - 20]];
    }
}

// ---------------------------------------------------------------------------
// Kernel 1: build WMMA B-operand fragments (register layout baked in) + ||w||^2
//   wfrag layout: [((i*6 + kk0)*12 + nt)*32 + lane] * 2 + v
//     B 4x16 f32 operand: VGPR v holds row k = v + 2*(lane>=16), col = lane&15
//   W2 layout:   [i*192 + ncol]
// ---------------------------------------------------------------------------
__global__ void weights_kernel(const float* __restrict__ ks,
                               const float* __restrict__ kr,
                               const float* __restrict__ kc,
                               float* __restrict__ wfrag,
                               float* __restrict__ W2) {
    int item = blockIdx.x * blockDim.x + threadIdx.x;
    if (item < 4 * NKSTEP * NTILE * 32) {            // 9216 B-fragment lanes
        int lane = item & 31;
        int rest = item >> 5;
        int nt   = rest % NTILE;  rest /= NTILE;
        int kk0  = rest % NKSTEP;
        int i    = rest / NKSTEP;
        int hi   = lane >> 4;
        int ncol = nt * 16 + (lane & 15);
        int o = ncol / 24, p = ncol % 24;
        for (int v = 0; v < 2; ++v) {
            int kk = kk0 * 4 + v + 2 * hi;
            wfrag[item * 2 + v] = wval(o, p, i, kk, ks, kr, kc);
        }
    } else if (item < 4 * NKSTEP * NTILE * 32 + 4 * NCOLS) {  // 768 norms
        int idx = item - 4 * NKSTEP * NTILE * 32;
        int i = idx / NCOLS, ncol = idx % NCOLS;
        int o = ncol / 24, p = ncol % 24;
        float s = 0.0f;
        for (int kk = 0; kk < KF; ++kk) {
            float w = wval(o, p, i, kk, ks, kr, kc);
            s += w * w;
        }
        W2[idx] = s;
    }
}

// ---------------------------------------------------------------------------
// Kernel 2: the WMMA distance GEMM + min-over-permutations pooling.
//   block = 128 threads = 4 waves; wave w handles in-channel i = w
//   blockIdx.x = M-tile (16 samples). 72x v_wmma_f32_16x16x4_f32 per wave.
//   pooled[b][o*4+i][s] = ||x||^2 + min_p ( ||w||^2 - 2 <x,w> )  -> sim_out
// ---------------------------------------------------------------------------
__global__ __launch_bounds__(128) void pool_kernel(
    const float* __restrict__ graph,   // (2,4,224,224)
    const float* __restrict__ types,   // (2,4,224)
    const float* __restrict__ wfrag,
    const float* __restrict__ W2,
    float* __restrict__ sim_out) {     // (2,32,NSUB) pre-softmax
    __shared__ float tile[4][16][NCOLS];   // 48 KB
    __shared__ float x2s[4][16];

    const int lane = threadIdx.x & 31;
    const int ch   = threadIdx.x >> 5;     // in-channel 0..3
    const int t    = blockIdx.x;
    const int hi   = lane >> 4;
    const int mrow = lane & 15;

    int n   = t * 16 + mrow;
    int ncl = n < NTOT ? n : NTOT - 1;     // clamp OOB rows (masked on store)
    int b   = ncl / NSUB;
    int s   = ncl % NSUB;
    int ir  = s / NH;                      // top-left row of subgraph
    int jc  = s % NH;                      // top-left col of subgraph

    const float* gch = graph + (size_t)(b * 4 + ch) * 224 * 224;
    const float* tch = types + (size_t)(b * 4 + ch) * 224;

    if (lane < 16) x2s[ch][lane] = 0.0f;
    __syncthreads();

    v8f acc[NTILE];
#pragma unroll
    for (int q = 0; q < NTILE; ++q) acc[q] = (v8f){};

#pragma unroll
    for (int kk0 = 0; kk0 < NKSTEP; ++kk0) {
        // A operand (16x4 f32): VGPR v holds k = v + 2*hi for row mrow
        v2f afrag;
#pragma unroll
        for (int v = 0; v < 2; ++v) {
            int kk = kk0 * 4 + v + 2 * hi;
            float val;
            if (kk < 16) {                          // struct: graph[ir+2a, jc+2b]
                int a = kk >> 2, bb = kk & 3;
                val = gch[(ir + 2 * a) * 224 + (jc + 2 * bb)];
            } else if (kk < 20) {                   // row types
                val = tch[ir + 2 * (kk - 16)];
            } else {                                // col types
                val = tch[jc + 2 * (kk - 20)];
            }
            afrag[v] = val;
            atomicAdd(&x2s[ch][mrow], val * val);   // ds_add_f32, ||x||^2
        }
#pragma unroll
        for (int nt = 0; nt < NTILE; ++nt) {
            const float* wp = wfrag + ((((ch * NKSTEP + kk0) * NTILE + nt) * 32 + lane) * 2);
            v2f bfrag;
            bfrag[0] = wp[0];
            bfrag[1] = wp[1];
            acc[nt] = __builtin_amdgcn_wmma_f32_16x16x4_f32(
                false, afrag, false, bfrag, (short)0, acc[nt], false, false);
        }
    }
    __syncthreads();

    // Spill D (16x192) to LDS as (||w||^2 - 2*cross); D layout: VGPR r ->
    // row r + 8*hi, col = lane&15 within each 16-wide tile.
    const int col = lane & 15;
#pragma unroll
    for (int nt = 0; nt < NTILE; ++nt) {
        int ncol  = nt * 16 + col;
        float w2v = W2[ch * NCOLS + ncol];
#pragma unroll
        for (int r = 0; r < 8; ++r) {
            tile[ch][r + 8 * hi][ncol] = w2v - 2.0f * acc[nt][r];
        }
    }
    __syncthreads();

    // Min over 24 permutations per output channel, add ||x||^2, scatter to
    // sim_out[b][o*4+ch][s] (softmax-friendly layout).
    for (int iter = 0; iter < 4; ++iter) {
        int idx  = iter * 32 + lane;       // 128 (row, o) pairs per wave
        int rowm = idx >> 3;
        int o    = idx & 7;
        float mn = tile[ch][rowm][o * 24];
#pragma unroll
        for (int p = 1; p < 24; ++p) mn = fminf(mn, tile[ch][rowm][o * 24 + p]);
        mn += x2s[ch][rowm];
        int n2 = t * 16 + rowm;
        if (n2 < NTOT) {
            int b2 = n2 / NSUB, s2 = n2 % NSUB;
            int c32 = o * 4 + ch;
            sim_out[(size_t)(b2 * 32 + c32) * NSUB + s2] = mn;
        }
    }
}

// ---------------------------------------------------------------------------
// Kernel 3: in-place per-row  (1 - softmax(x)) / NSUB ; 64 rows of 47524.
// ---------------------------------------------------------------------------
__global__ __launch_bounds__(256) void softmax_kernel(float* __restrict__ sim) {
    __shared__ float red[256];
    float* x = sim + (size_t)blockIdx.x * NSUB;
    int tid = threadIdx.x;

    float mx = -INFINITY;
    for (int s = tid; s < NSUB; s += 256) mx = fmaxf(mx, x[s]);
    red[tid] = mx;
    __syncthreads();
    for (int off = 128; off > 0; off >>= 1) {
        if (tid < off) red[tid] = fmaxf(red[tid], red[tid + off]);
        __syncthreads();
    }
    mx = red[0];
    __syncthreads();

    float sum = 0.0f;
    for (int s = tid; s < NSUB; s += 256) sum += expf(x[s] - mx);
    red[tid] = sum;
    __syncthreads();
    for (int off = 128; off > 0; off >>= 1) {
        if (tid < off) red[tid] += red[tid + off];
        __syncthreads();
    }
    sum = red[0];
    __syncthreads();

    const float inv   = 1.0f / sum;
    const float scale = 1.0f / (float)NSUB;
    for (int s = tid; s < NSUB; s += 256) {
        float e = expf(x[s] - mx) * inv;
        x[s] = (1.0f - e) * scale;
    }
}

// ---------------------------------------------------------------------------
// Kernel 4: h_mean[b,f] = (2/nH) * sum_m cnt[m] * features[b,m,f]
//   cnt[m] = #{(i,a) : i + 2a == m, 0<=i<nH, 0<=a<4}
// ---------------------------------------------------------------------------
__global__ void hmean_kernel(const float* __restrict__ feats, float* __restrict__ out) {
    int tid = blockIdx.x * blockDim.x + threadIdx.x;
    if (tid >= 2 * FDIM) return;
    int b = tid >> 7, f = tid & 127;
    float s = 0.0f;
    for (int m = 0; m < 224; ++m) {
        int cnt = 0;
#pragma unroll
        for (int a = 0; a < 4; ++a) {
            int i2 = m - 2 * a;
            cnt += (i2 >= 0 && i2 < NH) ? 1 : 0;
        }
        s += (float)cnt * feats[((size_t)b * 224 + m) * FDIM + f];
    }
    out[tid] = 2.0f * s / (float)NH;
}

// ---------------------------------------------------------------------------
extern "C" void kernel_launch(void* const* d_in, const int* in_sizes, int n_in,
                              void* d_out, int out_size, void* d_ws, size_t ws_size,
                              hipStream_t stream) {
    const float* graph    = (const float*)d_in[0];  // (2,4,224,224)
    const float* types    = (const float*)d_in[1];  // (2,4,224)
    const float* features = (const float*)d_in[2];  // (2,224,128)
    const float* k_struct = (const float*)d_in[3];  // (8,4,4,4)
    const float* k_row    = (const float*)d_in[4];  // (8,4,4)
    const float* k_col    = (const float*)d_in[5];  // (8,4,4)

    float* out   = (float*)d_out;        // [0,256): h_mean ; [256,...): out_sim
    float* wfrag = (float*)d_ws;         // 18432 floats
    float* W2    = wfrag + 4 * NKSTEP * NTILE * 32 * 2;  // +768 floats
    float* sim   = out + 2 * FDIM;       // (2,32,NSUB)

    int witems = 4 * NKSTEP * NTILE * 32 + 4 * NCOLS;  // 9984
    weights_kernel<<<(witems + 255) / 256, 256, 0, stream>>>(k_struct, k_row, k_col,
                                                             wfrag, W2);

    int mtiles = (NTOT + 15) / 16;       // 5941
    pool_kernel<<<mtiles, 128, 0, stream>>>(graph, types, wfrag, W2, sim);

    softmax_kernel<<<64, 256, 0, stream>>>(sim);

    hmean_kernel<<<1, 256, 0, stream>>>(features, out);
}